// CapsuleLayer_35596688949589
// MI455X (gfx1250) — compile-verified
//
#include <hip/hip_runtime.h>
#include <math.h>

typedef __attribute__((ext_vector_type(2))) float v2f;
typedef __attribute__((ext_vector_type(8))) float v8f;

#define B_SZ      32
#define IN_UNITS  2048
#define IN_CH     16
#define OUT_D     2048      // out_units*out_ch
#define OUT_U     64
#define OUT_C     32
#define XS_STRIDE 17        // LDS pad to kill bank conflicts on A loads

// ---------------------------------------------------------------------------
// Per-wave fp32 WMMA compute of u[b, nbase..nbase+31] for one in-unit i.
// A = x[:, i, :]  (32x16, staged in LDS, padded stride 17)
// B = W[i]        (16x2048, streamed from global)
// D = 2 (M-tiles) x 2 (N-tiles) f32 accumulators via V_WMMA_F32_16X16X4_F32.
// ---------------------------------------------------------------------------
__device__ __forceinline__ void compute_u(const float* __restrict__ Wi,
                                          const float* __restrict__ xs,
                                          int lane, int nbase, v8f u[2][2]) {
    const int hl  = lane >> 4;   // half-wave select (K pairing: lo lanes k,k+1; hi lanes k+2,k+3)
    const int l15 = lane & 15;

    v2f a[2][4];
#pragma unroll
    for (int mt = 0; mt < 2; ++mt)
#pragma unroll
        for (int kk = 0; kk < 4; ++kk) {
            const int b = mt * 16 + l15;
            const int k = kk * 4 + hl * 2;
            a[mt][kk].x = xs[b * XS_STRIDE + k];
            a[mt][kk].y = xs[b * XS_STRIDE + k + 1];
        }

    const v8f z = {0.f, 0.f, 0.f, 0.f, 0.f, 0.f, 0.f, 0.f};
#pragma unroll
    for (int mt = 0; mt < 2; ++mt)
#pragma unroll
        for (int nt = 0; nt < 2; ++nt) u[mt][nt] = z;

#pragma unroll
    for (int nt = 0; nt < 2; ++nt) {
        const int col = nbase + nt * 16 + l15;
#pragma unroll
        for (int kk = 0; kk < 4; ++kk) {
            const int k = kk * 4 + hl * 2;
            v2f bf;
            bf.x = Wi[(size_t)k * OUT_D + col];
            bf.y = Wi[(size_t)(k + 1) * OUT_D + col];
#pragma unroll
            for (int mt = 0; mt < 2; ++mt) {
                u[mt][nt] = __builtin_amdgcn_wmma_f32_16x16x4_f32(
                    false, a[mt][kk], false, bf, (short)0, u[mt][nt], false, false);
            }
        }
    }
}

// ---------------------------------------------------------------------------
// Pass 1: c = softmax(b_logits, axis=i).  One block per (b,o) pair.
// ---------------------------------------------------------------------------
__global__ void caps_softmax_kernel(const float* __restrict__ bl, float* __restrict__ c) {
    __shared__ float red[256];
    const int bo = blockIdx.x, b = bo >> 6, o = bo & 63;
    const int tid = threadIdx.x;
    const size_t base = (size_t)b * (IN_UNITS * OUT_U) + o;

    float vals[8];
    float mx = -1e30f;
#pragma unroll
    for (int r = 0; r < 8; ++r) {
        vals[r] = bl[base + (size_t)(tid + r * 256) * OUT_U];
        mx = fmaxf(mx, vals[r]);
    }
    red[tid] = mx; __syncthreads();
    for (int s = 128; s > 0; s >>= 1) { if (tid < s) red[tid] = fmaxf(red[tid], red[tid + s]); __syncthreads(); }
    mx = red[0]; __syncthreads();

    float sum = 0.f;
#pragma unroll
    for (int r = 0; r < 8; ++r) { vals[r] = expf(vals[r] - mx); sum += vals[r]; }
    red[tid] = sum; __syncthreads();
    for (int s = 128; s > 0; s >>= 1) { if (tid < s) red[tid] += red[tid + s]; __syncthreads(); }
    const float inv = 1.f / red[0];
#pragma unroll
    for (int r = 0; r < 8; ++r)
        c[base + (size_t)(tid + r * 256) * OUT_U] = vals[r] * inv;
}

// ---------------------------------------------------------------------------
// Pass 2: s_j partials.  Grid (8 n-chunks, 32 i-chunks), 256 thr (8 waves).
// Each wave: 32 columns = one out-unit o; recomputes u via WMMA, scales by
// c[b,i,o], accumulates over its 64 in-units, stores deterministic partials.
// ---------------------------------------------------------------------------
__global__ void caps_sj_kernel(const float* __restrict__ x, const float* __restrict__ W,
                               const float* __restrict__ c, float* __restrict__ s_part) {
    __shared__ float xs[B_SZ * XS_STRIDE];
    __shared__ float cs[256];                 // cs[b*8 + o_local]
    const int nchunk = blockIdx.x, ichunk = blockIdx.y;
    const int tid = threadIdx.x, wave = tid >> 5, lane = tid & 31;
    const int hl = lane >> 4, l15 = lane & 15;
    const int nbase = nchunk * 256 + wave * 32;
    const int obase = nchunk * 8;

    const v8f z = {0.f, 0.f, 0.f, 0.f, 0.f, 0.f, 0.f, 0.f};
    v8f sacc[2][2] = {{z, z}, {z, z}};

    for (int ii = 0; ii < 64; ++ii) {
        const int i = ichunk * 64 + ii;
        for (int t = tid; t < B_SZ * IN_CH; t += 256) {
            const int b = t >> 4, k = t & 15;
            xs[b * XS_STRIDE + k] = x[(size_t)b * (IN_UNITS * IN_CH) + (size_t)i * IN_CH + k];
        }
        {
            const int b = tid >> 3, ol = tid & 7;
            cs[tid] = c[(size_t)b * (IN_UNITS * OUT_U) + (size_t)i * OUT_U + obase + ol];
        }
        __syncthreads();

        v8f u[2][2];
        compute_u(W + (size_t)i * (IN_CH * OUT_D), xs, lane, nbase, u);

#pragma unroll
        for (int mt = 0; mt < 2; ++mt)
#pragma unroll
            for (int j = 0; j < 8; ++j) {
                const int b = mt * 16 + j + hl * 8;
                const float cc = cs[b * 8 + wave];
                sacc[mt][0][j] += cc * u[mt][0][j];
                sacc[mt][1][j] += cc * u[mt][1][j];
            }
        __syncthreads();
    }

#pragma unroll
    for (int mt = 0; mt < 2; ++mt)
#pragma unroll
        for (int nt = 0; nt < 2; ++nt)
#pragma unroll
            for (int j = 0; j < 8; ++j) {
                const int b = mt * 16 + j + hl * 8;
                const int n = nbase + nt * 16 + l15;
                s_part[(size_t)ichunk * (B_SZ * OUT_D) + (size_t)b * OUT_D + n] = sacc[mt][nt][j];
            }
}

// ---------------------------------------------------------------------------
// Pass 3: reduce partials over 32 i-chunks (fixed order) + squash.
// Grid 2048 blocks (b*64+o), 32 threads (d).
// ---------------------------------------------------------------------------
__global__ void caps_squash_kernel(const float* __restrict__ s_part, float* __restrict__ vout) {
    const int bo = blockIdx.x, b = bo >> 6, o = bo & 63;
    const int d = threadIdx.x;
    const int n = o * 32 + d;
    float s = 0.f;
    for (int ic = 0; ic < 32; ++ic)
        s += s_part[(size_t)ic * (B_SZ * OUT_D) + (size_t)b * OUT_D + n];
    float sq = s * s;
#pragma unroll
    for (int m = 16; m > 0; m >>= 1) sq += __shfl_xor(sq, m, 32);
    const float norm = sqrtf(sq);
    vout[(size_t)b * OUT_D + n] = s * (norm / (1.f + sq));   // s*norm/(1+norm^2)
}

// ---------------------------------------------------------------------------
// Pass 4: agreement update b_logits += <u[b,i,o,:], v[b,o,:]>.
// Same grid as pass 2; recomputes u via WMMA, dot over d via half-wave shfl.
// ---------------------------------------------------------------------------
__global__ void caps_update_kernel(const float* __restrict__ x, const float* __restrict__ W,
                                   const float* __restrict__ v, float* __restrict__ bl) {
    __shared__ float xs[B_SZ * XS_STRIDE];
    __shared__ float vs[B_SZ * 256];          // vs[b*256 + o_local*32 + d]
    const int nchunk = blockIdx.x, ichunk = blockIdx.y;
    const int tid = threadIdx.x, wave = tid >> 5, lane = tid & 31;
    const int hl = lane >> 4, l15 = lane & 15;
    const int nbase = nchunk * 256 + wave * 32;
    const int obase = nchunk * 8;
    const int o = obase + wave;

    for (int t = tid; t < B_SZ * 256; t += 256) {
        const int b = t >> 8, rem = t & 255;
        vs[t] = v[(size_t)b * OUT_D + obase * 32 + rem];
    }
    __syncthreads();

    for (int ii = 0; ii < 64; ++ii) {
        const int i = ichunk * 64 + ii;
        for (int t = tid; t < B_SZ * IN_CH; t += 256) {
            const int b = t >> 4, k = t & 15;
            xs[b * XS_STRIDE + k] = x[(size_t)b * (IN_UNITS * IN_CH) + (size_t)i * IN_CH + k];
        }
        __syncthreads();

        v8f u[2][2];
        compute_u(W + (size_t)i * (IN_CH * OUT_D), xs, lane, nbase, u);

#pragma unroll
        for (int mt = 0; mt < 2; ++mt)
#pragma unroll
            for (int j = 0; j < 8; ++j) {
                const int b = mt * 16 + j + hl * 8;
                float t0 = u[mt][0][j] * vs[b * 256 + wave * 32 + l15]
                         + u[mt][1][j] * vs[b * 256 + wave * 32 + 16 + l15];
                t0 += __shfl_xor(t0, 1, 32);
                t0 += __shfl_xor(t0, 2, 32);
                t0 += __shfl_xor(t0, 4, 32);
                t0 += __shfl_xor(t0, 8, 32);
                if (l15 == 0)
                    bl[(size_t)b * (IN_UNITS * OUT_U) + (size_t)i * OUT_U + o] += t0;
            }
        __syncthreads();
    }
}

// ---------------------------------------------------------------------------
extern "C" void kernel_launch(void* const* d_in, const int* in_sizes, int n_in,
                              void* d_out, int out_size, void* d_ws, size_t ws_size,
                              hipStream_t stream) {
    const float* x = (const float*)d_in[0];   // [32, 2048, 16]
    const float* W = (const float*)d_in[1];   // [2048, 16, 2048]
    float* out = (float*)d_out;               // [32, 64, 32]

    float* c  = (float*)d_ws;                       // 32*2048*64  = 4,194,304
    float* bl = c  + (size_t)B_SZ * IN_UNITS * OUT_U; // 4,194,304
    float* sp = bl + (size_t)B_SZ * IN_UNITS * OUT_U; // 32 * 65,536 = 2,097,152
    float* v  = sp + (size_t)32 * B_SZ * OUT_D;       // 65,536

    hipMemsetAsync(bl, 0, (size_t)B_SZ * IN_UNITS * OUT_U * sizeof(float), stream);

    const dim3 gridMM(8, 32), blk(256);
    for (int it = 0; it < 3; ++it) {
        caps_softmax_kernel<<<2048, 256, 0, stream>>>(bl, c);
        caps_sj_kernel<<<gridMM, blk, 0, stream>>>(x, W, c, sp);
        float* vt = (it == 2) ? out : v;
        caps_squash_kernel<<<2048, 32, 0, stream>>>(sp, vt);
        if (it < 2)
            caps_update_kernel<<<gridMM, blk, 0, stream>>>(x, W, vt, bl);
    }
}